// CBFRICL_28209345200463
// MI455X (gfx1250) — compile-verified
//
#include <hip/hip_runtime.h>
#include <math.h>

// ---------------- problem constants ----------------
#define N_USERS   50000
#define N_ITEMS   40000
#define DIM       64
#define NEDGE     1000000
#define NBLK      3
#define NLAY      2
#define BATCH     8192
#define N_NODES   (N_USERS + 1 + N_ITEMS + 1)   // 90002
#define NP        90016                          // padded to multiple of 16
#define NTILES    (NP / 16)                      // 5626

typedef __attribute__((ext_vector_type(2))) float v2f;
typedef __attribute__((ext_vector_type(8))) float v8f;

__device__ __forceinline__ void atomic_add_f32(float* p, float v) {
  __hip_atomic_fetch_add(p, v, __ATOMIC_RELAXED, __HIP_MEMORY_SCOPE_AGENT);
}

// ---------------- init: te = concat(user_emb, item_emb), pad rows zeroed ----
__global__ void k_init_te(const float* __restrict__ ue, const float* __restrict__ ie,
                          float* __restrict__ te) {
  size_t idx = (size_t)blockIdx.x * blockDim.x + threadIdx.x;
  const size_t UE  = (size_t)(N_USERS + 1) * DIM;
  const size_t TOT = (size_t)N_NODES * DIM;
  const size_t ALL = (size_t)NP * DIM;
  if (idx >= ALL) return;
  if (idx < UE)       te[idx] = ue[idx];
  else if (idx < TOT) te[idx] = ie[idx - UE];
  else                te[idx] = 0.0f;
}

__global__ void k_zero(float* __restrict__ p, size_t n) {
  size_t idx = (size_t)blockIdx.x * blockDim.x + threadIdx.x;
  if (idx < n) p[idx] = 0.0f;
}

// ---------------- degree count + symmetric norm ----------------
__global__ void k_deg(const int* __restrict__ col, float* __restrict__ deg) {
  size_t e = (size_t)blockIdx.x * blockDim.x + threadIdx.x;
  if (e >= NEDGE) return;
  atomic_add_f32(&deg[col[e]], 1.0f);
}

__global__ void k_dinv(float* __restrict__ deg) {
  size_t i = (size_t)blockIdx.x * blockDim.x + threadIdx.x;
  if (i >= NP) return;
  float d = deg[i];
  deg[i] = (d > 0.5f) ? rsqrtf(d) : 0.0f;   // max(deg,1)==deg when deg>=1
}

// ---------------- Y[n,d] = bias[d] (scatter target init, "out + b") ---------
__global__ void k_bias(float* __restrict__ Y, const float* __restrict__ b) {
  size_t idx = (size_t)blockIdx.x * blockDim.x + threadIdx.x;
  if (idx >= (size_t)NP * DIM) return;
  Y[idx] = b[idx & (DIM - 1)];
}

// ---------------- H = X @ W via v_wmma_f32_16x16x4_f32 ----------------------
// Block = 128 threads = 4 waves; blockIdx.x = M tile; wave id = N tile (0..3).
// A (16x4 f32): lanes 0-15 M=lane, VGPR0=K0,VGPR1=K1; lanes16-31 K2/K3.
// B (4x16 f32): same K mapping, N = lane%16.  C/D: VGPR v -> M = v + 8*half.
__global__ void __launch_bounds__(128) k_gemm64(const float* __restrict__ X,
                                                const float* __restrict__ W,
                                                float* __restrict__ H) {
  const int mt   = blockIdx.x;
  const int wv   = threadIdx.x >> 5;     // N tile (0..3)
  const int lane = threadIdx.x & 31;
  const int half = lane >> 4;            // 0: lanes 0-15, 1: lanes 16-31
  const int l16  = lane & 15;
  const int rowA = mt * 16 + l16;
  const int colB = wv * 16 + l16;
  const float* xr = X + (size_t)rowA * DIM;

  v8f c = {0.f, 0.f, 0.f, 0.f, 0.f, 0.f, 0.f, 0.f};
#pragma unroll
  for (int k = 0; k < DIM; k += 4) {
    const int kk = k + 2 * half;
    v2f a, b;
    a.x = xr[kk];
    a.y = xr[kk + 1];
    b.x = W[(size_t)kk * DIM + colB];
    b.y = W[(size_t)(kk + 1) * DIM + colB];
    c = __builtin_amdgcn_wmma_f32_16x16x4_f32(false, a, false, b,
                                              (short)0, c, false, false);
  }
#pragma unroll
  for (int v = 0; v < 8; ++v) {
    const int m = mt * 16 + v + 8 * half;
    H[(size_t)m * DIM + wv * 16 + l16] = c[v];
  }
}

// ---------------- msg scatter: Y[col] += H[row] * dinv[row]*dinv[col] -------
// 64 consecutive threads per edge -> coalesced 256B row read; float atomics.
__global__ void k_scatter(const int* __restrict__ row, const int* __restrict__ col,
                          const float* __restrict__ dinv,
                          const float* __restrict__ H, float* __restrict__ Y) {
  size_t t = (size_t)blockIdx.x * blockDim.x + threadIdx.x;
  size_t e = t >> 6;
  int    d = (int)(t & 63);
  if (e >= NEDGE) return;
  int r = row[e], c = col[e];
  float nrm = dinv[r] * dinv[c];
  if (nrm != 0.0f)
    atomic_add_f32(&Y[(size_t)c * DIM + d], H[(size_t)r * DIM + d] * nrm);
}

// ---------------- te += Y / max(||Y_row||, 1e-12)  (one wave per row) -------
__global__ void k_norm_accum(const float* __restrict__ Y, float* __restrict__ te) {
  const int wv   = threadIdx.x >> 5;
  const int lane = threadIdx.x & 31;
  const int r    = blockIdx.x * 8 + wv;
  if (r >= N_NODES) return;
  const float* yr = Y + (size_t)r * DIM;
  float x0 = yr[lane], x1 = yr[lane + 32];
  float ss = x0 * x0 + x1 * x1;
#pragma unroll
  for (int off = 16; off; off >>= 1) ss += __shfl_xor(ss, off, 32);
  float inv = 1.0f / fmaxf(sqrtf(ss), 1e-12f);
  float* tr = te + (size_t)r * DIM;
  tr[lane]      += x0 * inv;
  tr[lane + 32] += x1 * inv;
}

// ---------------- per-sample CL + BPR loss (one wave per sample) ------------
__global__ void k_loss(const float* __restrict__ te, const int* __restrict__ batch,
                       int bi, float* __restrict__ acc) {
  const int wv   = threadIdx.x >> 5;
  const int lane = threadIdx.x & 31;
  const int i    = blockIdx.x * 8 + wv;
  if (i >= BATCH) return;
  const int* bd = batch + ((size_t)i * NBLK + bi) * 3;
  const int u = bd[0];
  const int p = bd[1] + (N_USERS + 1);
  const int n = bd[2] + (N_USERS + 1);
  const float* uf = te + (size_t)u * DIM;
  const float* pf = te + (size_t)p * DIM;
  const float* nf = te + (size_t)n * DIM;
  float u0 = uf[lane], u1 = uf[lane + 32];
  float p0 = pf[lane], p1 = pf[lane + 32];
  float q0 = nf[lane], q1 = nf[lane + 32];
  float dup = u0 * p0 + u1 * p1;
  float dun = u0 * q0 + u1 * q1;
  float su  = u0 * u0 + u1 * u1;
  float sp  = p0 * p0 + p1 * p1;
  float sn  = q0 * q0 + q1 * q1;
#pragma unroll
  for (int off = 16; off; off >>= 1) {
    dup += __shfl_xor(dup, off, 32);
    dun += __shfl_xor(dun, off, 32);
    su  += __shfl_xor(su,  off, 32);
    sp  += __shfl_xor(sp,  off, 32);
    sn  += __shfl_xor(sn,  off, 32);
  }
  if (lane == 0) {
    float nu = sqrtf(su), npn = sqrtf(sp), nn = sqrtf(sn);
    float cosp = dup / fmaxf(nu * npn, 1e-8f);
    float cosn = dun / fmaxf(nu * nn,  1e-8f);
    // -log_softmax(logits)[0], logits = [cosp,cosn]/0.1 -> log1p(exp(l1-l0))
    float x  = (cosn - cosp) * 10.0f;
    float cl = (x > 0.0f) ? (x + log1pf(expf(-x))) : log1pf(expf(x));
    float s   = dup - dun;                      // raw-dot BPR scores
    float sig = 1.0f / (1.0f + expf(-s));
    float bpr = -logf(1e-10f + sig);
    atomic_add_f32(acc, (cl + bpr) * (1.0f / (float)BATCH));
  }
}

// ---------------- sum of squares reduction (regularizer) --------------------
__global__ void k_sqsum(const float* __restrict__ x, size_t n, float* __restrict__ acc) {
  size_t stride = (size_t)gridDim.x * blockDim.x;
  float s = 0.0f;
  for (size_t i = (size_t)blockIdx.x * blockDim.x + threadIdx.x; i < n; i += stride) {
    float v = x[i];
    s += v * v;
  }
#pragma unroll
  for (int off = 16; off; off >>= 1) s += __shfl_xor(s, off, 32);
  if ((threadIdx.x & 31) == 0) atomic_add_f32(acc, s);
}

// ---------------- finalize ---------------------------------------------------
__global__ void k_final(const float* __restrict__ sc, float* __restrict__ out) {
  // sc[0]=loss, sc[1]=sum_sq(user), sc[2]=sum_sq(item)
  out[0] = sc[0] + 0.01f * (sqrtf(sc[1]) + sqrtf(sc[2])) / (float)(N_ITEMS + 1);
}

// ============================================================================
extern "C" void kernel_launch(void* const* d_in, const int* in_sizes, int n_in,
                              void* d_out, int out_size, void* d_ws, size_t ws_size,
                              hipStream_t stream) {
  const float* user_emb = (const float*)d_in[0];   // (50001, 64)
  const float* item_emb = (const float*)d_in[1];   // (40001, 64)
  const float* gcn_w    = (const float*)d_in[2];   // (3, 2, 64, 64)
  const float* gcn_b    = (const float*)d_in[3];   // (3, 2, 64)
  const int*   edge     = (const int*)  d_in[4];   // (3, 2, 1e6)
  const int*   batch    = (const int*)  d_in[5];   // (8192, 3, 3)
  float*       out      = (float*)d_out;

  // workspace layout (floats)
  float* te   = (float*)d_ws;                 // NP*64
  float* bufA = te   + (size_t)NP * DIM;      // NP*64  (layer output)
  float* H    = bufA + (size_t)NP * DIM;      // NP*64  (GEMM result)
  float* dinv = H    + (size_t)NP * DIM;      // NP
  float* sc   = dinv + (size_t)NP;            // scalars: loss, sqU, sqI

  const size_t rowelems = (size_t)NP * DIM;   // 5,761,024
  const int    gElem    = (int)((rowelems + 255) / 256);

  // te = concat(embeddings); zero scalar accumulators
  k_init_te<<<gElem, 256, 0, stream>>>(user_emb, item_emb, te);
  k_zero<<<1, 32, 0, stream>>>(sc, 3);

  // regularizer sums (independent of GCN)
  k_sqsum<<<2048, 256, 0, stream>>>(user_emb, (size_t)(N_USERS + 1) * DIM, sc + 1);
  k_sqsum<<<2048, 256, 0, stream>>>(item_emb, (size_t)(N_ITEMS + 1) * DIM, sc + 2);

  const int gScatter = (int)(((size_t)NEDGE * 64) / 256);  // 250000
  const int gDeg     = (NEDGE + 255) / 256;
  const int gDinv    = (NP + 255) / 256;
  const int gNorm    = (N_NODES + 7) / 8;
  const int gLoss    = (BATCH + 7) / 8;

  for (int bi = 0; bi < NBLK; ++bi) {
    const int* rowp = edge + (size_t)bi * 2 * NEDGE;
    const int* colp = rowp + NEDGE;

    // symmetric normalization (same for both layers of this block)
    k_zero<<<gDinv, 256, 0, stream>>>(dinv, NP);
    k_deg<<<gDeg, 256, 0, stream>>>(colp, dinv);
    k_dinv<<<gDinv, 256, 0, stream>>>(dinv);

    // ---- layer 0: H = te @ W0 ; bufA = bias0 ; scatter(H -> bufA)
    const float* W0 = gcn_w + ((size_t)bi * NLAY + 0) * DIM * DIM;
    const float* b0 = gcn_b + ((size_t)bi * NLAY + 0) * DIM;
    k_gemm64<<<NTILES, 128, 0, stream>>>(te, W0, H);
    k_bias<<<gElem, 256, 0, stream>>>(bufA, b0);
    k_scatter<<<gScatter, 256, 0, stream>>>(rowp, colp, dinv, H, bufA);

    // ---- layer 1: H = bufA @ W1 ; bufA = bias1 ; scatter(H -> bufA)
    const float* W1 = gcn_w + ((size_t)bi * NLAY + 1) * DIM * DIM;
    const float* b1 = gcn_b + ((size_t)bi * NLAY + 1) * DIM;
    k_gemm64<<<NTILES, 128, 0, stream>>>(bufA, W1, H);
    k_bias<<<gElem, 256, 0, stream>>>(bufA, b1);
    k_scatter<<<gScatter, 256, 0, stream>>>(rowp, colp, dinv, H, bufA);

    // ---- te += normalize(bufA); then losses on updated te
    k_norm_accum<<<gNorm, 256, 0, stream>>>(bufA, te);
    k_loss<<<gLoss, 256, 0, stream>>>(te, batch, bi, sc);
  }

  k_final<<<1, 1, 0, stream>>>(sc, out);
  (void)in_sizes; (void)n_in; (void)out_size; (void)ws_size;
}